// GATv2Encoder_5660766896561
// MI455X (gfx1250) — compile-verified
//
#include <hip/hip_runtime.h>
#include <hip/hip_bf16.h>

// ---------------------------------------------------------------------------
// GATv2 encoder for gfx1250 (MI455X).
// Dense node transforms use V_WMMA_F32_16X16X4_F32 (exact fp32 matrix math).
// Edge phase is warp-per-edge (wave32): 32 lanes x 4 channels = 128 channels.
// ---------------------------------------------------------------------------

typedef float v2f __attribute__((ext_vector_type(2)));
typedef float v8f __attribute__((ext_vector_type(8)));

#define HIDC 128   // channel width everywhere (IN == HID == 128)

// ---------------- utility -------------------------------------------------

__global__ void zero_f32_kernel(float* __restrict__ p, int n) {
    int i = blockIdx.x * blockDim.x + threadIdx.x;
    if (i < n) p[i] = 0.0f;
}

// order-preserving float->u32 map (for atomic max over signed floats)
__device__ __forceinline__ unsigned fenc(float f) {
    unsigned u = __float_as_uint(f);
    return (u & 0x80000000u) ? ~u : (u | 0x80000000u);
}

// ---------------- GEMM: O[M,128] = A[M,128] @ W[128,128] -------------------
// One wave per 16x16 output tile. blockDim = 256 (8 waves), wave w -> tileN=w,
// blockIdx.x -> tileM. M must be a multiple of 16 (50000 = 3125*16).
// A-matrix 16x4 f32 layout: lanes 0-15 rows M=0..15 hold K=k,k+1 in v0,v1;
// lanes 16-31 hold K=k+2,k+3. B 4x16 mirrors with N across lanes.
__global__ void gemm128_wmma_kernel(const float* __restrict__ A,
                                    const float* __restrict__ W,
                                    float* __restrict__ O, int M) {
    const int lane    = threadIdx.x & 31;
    const int wave    = threadIdx.x >> 5;      // tileN 0..7
    const int m0      = blockIdx.x * 16;
    const int halfsel = lane >> 4;             // 0 | 1
    const int l15     = lane & 15;
    const int koff    = halfsel * 2;
    const int col     = wave * 16 + l15;

    const float* arow = A + (size_t)(m0 + l15) * HIDC + koff;
    const float* wcol = W + col;

    v8f c = {0.f, 0.f, 0.f, 0.f, 0.f, 0.f, 0.f, 0.f};
#pragma unroll
    for (int k = 0; k < HIDC; k += 4) {
        v2f a, b;
        a[0] = arow[k];
        a[1] = arow[k + 1];
        b[0] = wcol[(size_t)(k + koff) * HIDC];
        b[1] = wcol[(size_t)(k + koff + 1) * HIDC];
        c = __builtin_amdgcn_wmma_f32_16x16x4_f32(
                /*neg_a=*/false, a, /*neg_b=*/false, b,
                /*c_mod=*/(short)0, c, /*reuse_a=*/false, /*reuse_b=*/false);
    }
    // C/D layout: VGPR v: lanes0-15 -> row v, lanes16-31 -> row v+8; N = lane&15
    const int r0 = m0 + halfsel * 8;
#pragma unroll
    for (int v = 0; v < 8; ++v)
        O[(size_t)(r0 + v) * HIDC + col] = c[v];
}

// ---------------- edge logits + segment max --------------------------------
// warp per edge; lane handles channels c = i*32+lane, i=0..3.
// For HEADS==4 (head_dim 32) head == i; for HEADS==1 all channels -> head 0.
template <int HEADS>
__global__ void edge_logit_kernel(const float* __restrict__ XL,
                                  const float* __restrict__ XR,
                                  const int* __restrict__ src,
                                  const int* __restrict__ dst,
                                  const float* __restrict__ eattr,
                                  const float* __restrict__ We,   // [2,128]
                                  const float* __restrict__ att,  // [HEADS*HD]=128
                                  float* __restrict__ logit,      // [E,HEADS]
                                  unsigned* __restrict__ mx,      // [N,HEADS]
                                  int E) {
    int e    = (int)(((size_t)blockIdx.x * blockDim.x + threadIdx.x) >> 5);
    int lane = threadIdx.x & 31;
    if (e >= E) return;
    int s = src[e], d = dst[e];
    float e0 = eattr[2 * (size_t)e], e1 = eattr[2 * (size_t)e + 1];

    float acc[HEADS];
#pragma unroll
    for (int h = 0; h < HEADS; ++h) acc[h] = 0.f;

#pragma unroll
    for (int i = 0; i < 4; ++i) {
        int c = i * 32 + lane;
        float m = XL[(size_t)s * HIDC + c] + XR[(size_t)d * HIDC + c]
                + e0 * We[c] + e1 * We[HIDC + c];
        float sl = (m > 0.f) ? m : 0.2f * m;  // leaky_relu(0.2)
        acc[(HEADS == 4) ? i : 0] += sl * att[c];
    }
#pragma unroll
    for (int h = 0; h < HEADS; ++h) {
        float v = acc[h];
#pragma unroll
        for (int off = 16; off >= 1; off >>= 1)
            v += __shfl_xor(v, off, 32);
        if (lane == 0) {
            logit[(size_t)e * HEADS + h] = v;
            atomicMax(&mx[(size_t)d * HEADS + h], fenc(v));
        }
    }
}

// decode order-preserving u32 back to float, in place
__global__ void mx_decode_kernel(unsigned* __restrict__ mx, int n) {
    int i = blockIdx.x * blockDim.x + threadIdx.x;
    if (i >= n) return;
    unsigned u = mx[i];
    unsigned b = (u & 0x80000000u) ? (u & 0x7FFFFFFFu) : ~u;
    ((float*)mx)[i] = __uint_as_float(b);
}

// ex = exp(logit - mx[dst]); den[dst] += ex  (overwrites logit with ex)
template <int HEADS>
__global__ void edge_exp_kernel(float* __restrict__ logit,
                                const int* __restrict__ dst,
                                const float* __restrict__ mxf,
                                float* __restrict__ den, int E) {
    size_t t = (size_t)blockIdx.x * blockDim.x + threadIdx.x;
    if (t >= (size_t)E * HEADS) return;
    int e = (int)(t / HEADS);
    int h = (int)(t % HEADS);
    int d = dst[e];
    float ex = __expf(logit[t] - mxf[(size_t)d * HEADS + h]);
    logit[t] = ex;
    atomicAdd(&den[(size_t)d * HEADS + h], ex);
}

// out[dst] += xl[src] * alpha   (warp per edge)
template <int HEADS>
__global__ void edge_aggr_kernel(const float* __restrict__ XL,
                                 const int* __restrict__ src,
                                 const int* __restrict__ dst,
                                 const float* __restrict__ ex,
                                 const float* __restrict__ den,
                                 float* __restrict__ out, int E) {
    int e    = (int)(((size_t)blockIdx.x * blockDim.x + threadIdx.x) >> 5);
    int lane = threadIdx.x & 31;
    if (e >= E) return;
    int s = src[e], d = dst[e];
    float al[HEADS];
#pragma unroll
    for (int h = 0; h < HEADS; ++h)
        al[h] = ex[(size_t)e * HEADS + h] / (den[(size_t)d * HEADS + h] + 1e-16f);
#pragma unroll
    for (int i = 0; i < 4; ++i) {
        int c = i * 32 + lane;
        atomicAdd(&out[(size_t)d * HIDC + c],
                  XL[(size_t)s * HIDC + c] * al[(HEADS == 4) ? i : 0]);
    }
}

// h = BN(ELU(acc + bias));  BN eval: g/sqrt(1+eps) * v + b
__global__ void node_post_kernel(const float* __restrict__ acc,
                                 const float* __restrict__ bias,
                                 const float* __restrict__ g,
                                 const float* __restrict__ b,
                                 float* __restrict__ hout, int n) {
    int i = blockIdx.x * blockDim.x + threadIdx.x;
    if (i >= n) return;
    int c = i & (HIDC - 1);
    float v = acc[i] + bias[c];
    v = (v > 0.f) ? v : (__expf(v) - 1.f);                 // ELU
    const float inv = 0.99999500003749969f;                // 1/sqrt(1+1e-5)
    hout[i] = v * (g[c] * inv) + b[c];
}

// ---------------- global mean pool -----------------------------------------

__global__ void pool_accum_kernel(const float* __restrict__ h,
                                  const int* __restrict__ batch,
                                  float* __restrict__ sums,
                                  float* __restrict__ cnt, int N) {
    int i = blockIdx.x * blockDim.x + threadIdx.x;
    if (i >= N * HIDC) return;
    int nidx = i >> 7;
    int c    = i & (HIDC - 1);
    int gidx = batch[nidx];
    atomicAdd(&sums[(size_t)gidx * HIDC + c], h[i]);
    if (c == 0) atomicAdd(&cnt[gidx], 1.0f);
}

__global__ void pool_div_kernel(const float* __restrict__ sums,
                                const float* __restrict__ cnt,
                                float* __restrict__ out, int n) {
    int i = blockIdx.x * blockDim.x + threadIdx.x;
    if (i >= n) return;
    out[i] = sums[i] / fmaxf(cnt[i >> 7], 1.0f);
}

// ---------------- layer orchestration --------------------------------------

static void launch_layer(const float* Hin, const float* Wl, const float* Wr,
                         const float* We, const float* att, const float* bias,
                         const float* bng, const float* bnb, int heads,
                         const int* src, const int* dst, const float* eattr,
                         float* XL, float* XR, float* OUT, float* EX,
                         unsigned* MX, float* DEN, float* Hout,
                         int N, int E, hipStream_t stream) {
    const dim3 blk(256);
    // dense transforms (WMMA f32)
    gemm128_wmma_kernel<<<N / 16, blk, 0, stream>>>(Hin, Wl, XL, N);
    gemm128_wmma_kernel<<<N / 16, blk, 0, stream>>>(Hin, Wr, XR, N);
    // clear accumulators
    int nh = N * heads;
    zero_f32_kernel<<<(nh + 255) / 256, blk, 0, stream>>>((float*)MX, nh);
    zero_f32_kernel<<<(nh + 255) / 256, blk, 0, stream>>>(DEN, nh);
    zero_f32_kernel<<<(N * HIDC + 255) / 256, blk, 0, stream>>>(OUT, N * HIDC);
    // edge phase
    int eblocks = (int)(((size_t)E * 32 + 255) / 256);
    size_t eh = (size_t)E * heads;
    int ehblocks = (int)((eh + 255) / 256);
    if (heads == 4) {
        edge_logit_kernel<4><<<eblocks, blk, 0, stream>>>(XL, XR, src, dst, eattr, We, att, EX, MX, E);
        mx_decode_kernel<<<(nh + 255) / 256, blk, 0, stream>>>(MX, nh);
        edge_exp_kernel<4><<<ehblocks, blk, 0, stream>>>(EX, dst, (const float*)MX, DEN, E);
        edge_aggr_kernel<4><<<eblocks, blk, 0, stream>>>(XL, src, dst, EX, DEN, OUT, E);
    } else {
        edge_logit_kernel<1><<<eblocks, blk, 0, stream>>>(XL, XR, src, dst, eattr, We, att, EX, MX, E);
        mx_decode_kernel<<<(nh + 255) / 256, blk, 0, stream>>>(MX, nh);
        edge_exp_kernel<1><<<ehblocks, blk, 0, stream>>>(EX, dst, (const float*)MX, DEN, E);
        edge_aggr_kernel<1><<<eblocks, blk, 0, stream>>>(XL, src, dst, EX, DEN, OUT, E);
    }
    // bias + ELU + BN
    node_post_kernel<<<(N * HIDC + 255) / 256, blk, 0, stream>>>(OUT, bias, bng, bnb, Hout, N * HIDC);
}

// ---------------- entry ----------------------------------------------------

extern "C" void kernel_launch(void* const* d_in, const int* in_sizes, int n_in,
                              void* d_out, int out_size, void* d_ws, size_t ws_size,
                              hipStream_t stream) {
    const float* x     = (const float*)d_in[0];
    const int*   ei    = (const int*)d_in[1];
    const float* eattr = (const float*)d_in[2];
    const int*   batch = (const int*)d_in[3];

    const int N  = in_sizes[0] / HIDC;  // 50000
    const int E  = in_sizes[1] / 2;     // 1600000
    const int NG = 64;

    const int* src = ei;
    const int* dst = ei + E;

    // parameter pointers: support both dict-insertion order and jax-sorted order
    const float *Wl[3], *Wr[3], *We[3], *att[3], *bias[3], *g[3], *b[3];
    if (in_sizes[4] == HIDC * HIDC) {
        // insertion order: l1..l3 {Wl,Wr,We,att,bias}, then bn1..bn3 {g,b}
        int idx = 4;
        for (int i = 0; i < 3; ++i) {
            Wl[i]   = (const float*)d_in[idx++];
            Wr[i]   = (const float*)d_in[idx++];
            We[i]   = (const float*)d_in[idx++];
            att[i]  = (const float*)d_in[idx++];
            bias[i] = (const float*)d_in[idx++];
        }
        for (int i = 0; i < 3; ++i) {
            g[i] = (const float*)d_in[idx++];
            b[i] = (const float*)d_in[idx++];
        }
    } else {
        // jax tree order (sorted keys): bn1{b,g}..bn3, l1{We,Wl,Wr,att,bias}..l3
        int idx = 4;
        for (int i = 0; i < 3; ++i) {
            b[i] = (const float*)d_in[idx++];
            g[i] = (const float*)d_in[idx++];
        }
        for (int i = 0; i < 3; ++i) {
            We[i]   = (const float*)d_in[idx++];
            Wl[i]   = (const float*)d_in[idx++];
            Wr[i]   = (const float*)d_in[idx++];
            att[i]  = (const float*)d_in[idx++];
            bias[i] = (const float*)d_in[idx++];
        }
    }

    // workspace carve-up (fp32)
    float* ws   = (float*)d_ws;
    float* H    = ws;  ws += (size_t)N * HIDC;
    float* XL   = ws;  ws += (size_t)N * HIDC;
    float* XR   = ws;  ws += (size_t)N * HIDC;
    float* OUT  = ws;  ws += (size_t)N * HIDC;
    float* EX   = ws;  ws += (size_t)E * 4;
    unsigned* MX = (unsigned*)ws;  ws += (size_t)N * 4;
    float* DEN  = ws;  ws += (size_t)N * 4;
    float* SUMS = ws;  ws += (size_t)NG * HIDC;
    float* CNT  = ws;  ws += NG;

    // layer 1: input x, heads=4, concat
    launch_layer(x, Wl[0], Wr[0], We[0], att[0], bias[0], g[0], b[0], 4,
                 src, dst, eattr, XL, XR, OUT, EX, MX, DEN, H, N, E, stream);
    // layer 2: heads=4, concat (H -> H in place is safe: H read only by GEMMs)
    launch_layer(H, Wl[1], Wr[1], We[1], att[1], bias[1], g[1], b[1], 4,
                 src, dst, eattr, XL, XR, OUT, EX, MX, DEN, H, N, E, stream);
    // layer 3: heads=1, out_ch=128 (mean over 1 head == identity)
    launch_layer(H, Wl[2], Wr[2], We[2], att[2], bias[2], g[2], b[2], 1,
                 src, dst, eattr, XL, XR, OUT, EX, MX, DEN, H, N, E, stream);

    // global mean pool
    const dim3 blk(256);
    zero_f32_kernel<<<(NG * HIDC + 255) / 256, blk, 0, stream>>>(SUMS, NG * HIDC);
    zero_f32_kernel<<<(NG + 255) / 256, blk, 0, stream>>>(CNT, NG);
    pool_accum_kernel<<<(N * HIDC + 255) / 256, blk, 0, stream>>>(H, batch, SUMS, CNT, N);
    pool_div_kernel<<<(NG * HIDC + 255) / 256, blk, 0, stream>>>(SUMS, CNT, (float*)d_out, NG * HIDC);
}